// KSpaceMap_2576980378166
// MI455X (gfx1250) — compile-verified
//
#include <hip/hip_runtime.h>
#include <math.h>

typedef float v2f __attribute__((ext_vector_type(2)));
typedef float v8f __attribute__((ext_vector_type(8)));

#define WIMG 128
#define TILE_M 16
#define NTHREADS 512

__global__ __launch_bounds__(NTHREADS)
void kspace_map_kernel(const float* __restrict__ x, float* __restrict__ out) {
    // LDS: input tile, P/Q (row-DFT real/imag parts), twiddle tables
    __shared__ float xs[TILE_M * WIMG];   // x[m][w'] for 16 rows
    __shared__ float Ps[TILE_M * WIMG];   // P[m][i]
    __shared__ float Qs[TILE_M * WIMG];   // Q[m][i]
    __shared__ float tc[WIMG];            // cos(2*pi*k/128)
    __shared__ float ts[WIMG];            // sin(2*pi*k/128)

    const int t    = threadIdx.x;
    const int tile = blockIdx.x;          // row-tile: rows r = tile*16 + m, r = b*128 + h

    // Twiddle tables (double precision eval, 128 entries, negligible cost)
    if (t < WIMG) {
        double ang = (6.283185307179586476925286766559 / (double)WIMG) * (double)t;
        tc[t] = (float)cos(ang);
        ts[t] = (float)sin(ang);
    }
    // Load contiguous 16x128 input tile
    const float* src = x + (size_t)tile * (TILE_M * WIMG);
    for (int j = t; j < TILE_M * WIMG; j += NTHREADS) xs[j] = src[j];
    __syncthreads();

    const int wv   = t >> 5;       // wave 0..15
    const int lane = t & 31;
    const int half = lane >> 4;    // 0/1
    const int mn   = lane & 15;    // M (for A/D) and N (for B/D)

    // ---- Phase 1: P/Q[m][i] = sum_k xs[m][k] * cos/sin(2*pi*i*k/W) ----
    // 16 wave-jobs: col-tile ct = wv&7 (i0 = 16*ct), P for wv<8, Q for wv>=8.
    {
        const int  i0  = (wv & 7) * 16;
        const bool doP = (wv < 8);
        const float* tab = doP ? tc : ts;
        const int  i   = i0 + mn;              // B-fragment column
        v8f acc = {};
        for (int ks = 0; ks < 32; ++ks) {
            const int k0 = ks * 4 + 2 * half;  // per-lane K pair
            v2f a, b;
            a.x = xs[mn * WIMG + k0];
            a.y = xs[mn * WIMG + k0 + 1];
            b.x = tab[(i * k0) & (WIMG - 1)];
            b.y = tab[(i * (k0 + 1)) & (WIMG - 1)];
            // D = A(16x4) * B(4x16) + C, fp32 exact
            acc = __builtin_amdgcn_wmma_f32_16x16x4_f32(
                      false, a, false, b, (short)0, acc, false, false);
        }
        float* dst = doP ? Ps : Qs;
        #pragma unroll
        for (int j = 0; j < 8; ++j)            // C layout: M = j + 8*half, N = mn
            dst[(j + 8 * half) * WIMG + i0 + mn] = acc[j];
    }
    __syncthreads();

    // ---- Phase 2: out[b,i,h,w] = (cos(2pi*i*w/W)*P + sin(2pi*i*w/W)*Q)/W ----
    // Wave wv owns row m = wv; lanes cover w = 4*lane..4*lane+3 -> coalesced b128 stores.
    const int m = wv;
    const int r = tile * TILE_M + m;
    const int b = r >> 7;
    const int h = r & 127;
    const int w0 = lane * 4;
    float* obase = out + ((size_t)b << 21) + ((size_t)h << 7) + (size_t)w0;
    const float scale = 1.0f / (float)WIMG;

    for (int i = 0; i < WIMG; ++i) {
        const float P = Ps[m * WIMG + i] * scale;
        const float Q = Qs[m * WIMG + i] * scale;
        int idx = (i * w0) & (WIMG - 1);
        float4 v;
        v.x = P * tc[idx] + Q * ts[idx]; idx = (idx + i) & (WIMG - 1);
        v.y = P * tc[idx] + Q * ts[idx]; idx = (idx + i) & (WIMG - 1);
        v.z = P * tc[idx] + Q * ts[idx]; idx = (idx + i) & (WIMG - 1);
        v.w = P * tc[idx] + Q * ts[idx];
        *reinterpret_cast<float4*>(obase + ((size_t)i << 14)) = v;
    }
}

extern "C" void kernel_launch(void* const* d_in, const int* in_sizes, int n_in,
                              void* d_out, int out_size, void* d_ws, size_t ws_size,
                              hipStream_t stream) {
    const float* x = (const float*)d_in[0];   // (16,1,128,128) float32; mask d_in[1] unused
    float* out = (float*)d_out;               // (16,128,128,128) float32
    // 2048 (b,h) rows / 16 per block = 128 blocks, 512 threads (16 waves)
    kspace_map_kernel<<<128, NTHREADS, 0, stream>>>(x, out);
}